// HashGridEncoder_64072322121975
// MI455X (gfx1250) — compile-verified
//
#include <hip/hip_runtime.h>
#include <stdint.h>

typedef uint32_t u32;
typedef __attribute__((ext_vector_type(4))) u32   v4u;
typedef __attribute__((ext_vector_type(8))) int   v8i;
typedef __attribute__((ext_vector_type(4))) int   v4i;
typedef __attribute__((ext_vector_type(2))) float v2f;
typedef __attribute__((ext_vector_type(4))) float v4f;

#define NLEV 16
#define TBL   (1u << 19)
#define TMASK (TBL - 1u)

__device__ __forceinline__ u32 umin2(u32 a, u32 b) { return a < b ? a : b; }

__global__ __launch_bounds__(256) void hashgrid_fwd(
    const float* __restrict__ xin,
    const float* __restrict__ table,
    float* __restrict__ out,
    int npts)
{
  // per-level resolutions: floor(16 * (2048/16)^(l/15))
  constexpr int kRes[NLEV] = {16, 22, 30, 42, 58, 80, 111, 153,
                              212, 294, 406, 561, 776, 1072, 1482, 2048};

  // level-0 dense table: 16^3 entries x 2 floats = 32 KB, staged via TDM
  __shared__ float tab0[4096 * 2];

  // ---- Tensor Data Mover: stage level-0 table into LDS (wave 0 issues) ----
  if (threadIdx.x < 32u) {
    u32 lds_addr    = (u32)(uintptr_t)(&tab0[0]);
    uint64_t gaddr  = (uint64_t)(uintptr_t)table;

    // D# group 0: count=1 | lds_addr | global_addr[56:0] | type=2
    v4u g0;
    g0.x = 1u;
    g0.y = lds_addr;
    g0.z = (u32)gaddr;
    g0.w = ((u32)(gaddr >> 32) & 0x01FFFFFFu) | (2u << 30);

    // D# group 1: wg_mask=0, data_size=3 (8B), 1-D tile of 4096 elements
    v8i g1;
    g1[0] = 0x00030000;            // data_size=3 at bits[17:16]
    g1[1] = (int)(4096u << 16);    // tensor_dim0[15:0] = 4096
    g1[2] = (int)(1u << 16);       // tensor_dim0[31:16]=0, tensor_dim1=1
    g1[3] = (int)(4096u << 16);    // tile_dim0 = 4096
    g1[4] = 0;                     // tile_dim1=0 (unused), tile_dim2=0
    g1[5] = 4096;                  // tensor_dim0_stride[31:0]
    g1[6] = 0;
    g1[7] = 0;

    v4i g2; g2[0] = 1; g2[1] = 1; g2[2] = 0; g2[3] = 0;  // dim2/dim3 = 1
    v4i g3; g3[0] = 0; g3[1] = 0; g3[2] = 0; g3[3] = 0;
    v8i g4; g4[0] = 0; g4[1] = 0; g4[2] = 0; g4[3] = 0;
            g4[4] = 0; g4[5] = 0; g4[6] = 0; g4[7] = 0;

    // 6-arg form (amdgpu-toolchain / clang-23)
    __builtin_amdgcn_tensor_load_to_lds(g0, g1, g2, g3, g4, 0);
    __builtin_amdgcn_s_wait_tensorcnt(0);
  }
  __syncthreads();

  int n = (int)(blockIdx.x * blockDim.x + threadIdx.x);
  if (n >= npts) return;

  float ux = (xin[3 * n + 0] + 1.0f) * 0.5f;
  float uy = (xin[3 * n + 1] + 1.0f) * 0.5f;
  float uz = (xin[3 * n + 2] + 1.0f) * 0.5f;

  float acc[2 * NLEV];

  #pragma unroll
  for (int l = 0; l < NLEV; ++l) {
    const int   R   = kRes[l];
    const u32   Rm1 = (u32)(R - 1);
    const float sc  = (float)(R - 1);

    float fx = ux * sc, fy = uy * sc, fz = uz * sc;
    float gx = floorf(fx), gy = floorf(fy), gz = floorf(fz);
    float tx = fx - gx, ty = fy - gy, tz = fz - gz;
    u32 cx = (u32)gx, cy = (u32)gy, cz = (u32)gz;
    u32 cx1 = umin2(cx + 1u, Rm1);
    u32 cy1 = umin2(cy + 1u, Rm1);
    u32 cz1 = umin2(cz + 1u, Rm1);

    // corner order: bit2 = x-offset, bit1 = y-offset, bit0 = z-offset
    u32 idx[8];
    if ((long long)R * R * R <= (long long)TBL) {
      // dense (non-colliding) index for coarse levels
      u32 ax0 = cx,               ax1 = cx1;
      u32 ay0 = cy  * (u32)R,     ay1 = cy1 * (u32)R;
      u32 az0 = cz  * (u32)(R*R), az1 = cz1 * (u32)(R*R);
      idx[0] = ax0 + ay0 + az0;  idx[1] = ax0 + ay0 + az1;
      idx[2] = ax0 + ay1 + az0;  idx[3] = ax0 + ay1 + az1;
      idx[4] = ax1 + ay0 + az0;  idx[5] = ax1 + ay0 + az1;
      idx[6] = ax1 + ay1 + az0;  idx[7] = ax1 + ay1 + az1;
    } else {
      // spatial hash: xor of coord*prime, masked to 2^19
      u32 hx0 = cx,                 hx1 = cx1;
      u32 hy0 = cy  * 2654435761u,  hy1 = cy1 * 2654435761u;
      u32 hz0 = cz  * 805459861u,   hz1 = cz1 * 805459861u;
      idx[0] = (hx0 ^ hy0 ^ hz0) & TMASK;  idx[1] = (hx0 ^ hy0 ^ hz1) & TMASK;
      idx[2] = (hx0 ^ hy1 ^ hz0) & TMASK;  idx[3] = (hx0 ^ hy1 ^ hz1) & TMASK;
      idx[4] = (hx1 ^ hy0 ^ hz0) & TMASK;  idx[5] = (hx1 ^ hy0 ^ hz1) & TMASK;
      idx[6] = (hx1 ^ hy1 ^ hz0) & TMASK;  idx[7] = (hx1 ^ hy1 ^ hz1) & TMASK;
    }

    // trilinear weights matching corner order
    float wx0 = 1.0f - tx, wx1 = tx;
    float wy0 = 1.0f - ty, wy1 = ty;
    float wz0 = 1.0f - tz, wz1 = tz;
    float wjk0 = wy0 * wz0, wjk1 = wy0 * wz1, wjk2 = wy1 * wz0, wjk3 = wy1 * wz1;
    float w[8] = { wx0 * wjk0, wx0 * wjk1, wx0 * wjk2, wx0 * wjk3,
                   wx1 * wjk0, wx1 * wjk1, wx1 * wjk2, wx1 * wjk3 };

    // gather 8 feature pairs: level 0 from LDS, others from L2-resident table
    v2f f[8];
    if (l == 0) {
      #pragma unroll
      for (int c = 0; c < 8; ++c)
        f[c] = *(const v2f*)&tab0[2u * idx[c]];
    } else {
      const v2f* tl = (const v2f*)(table + (size_t)l * (size_t)(TBL * 2u));
      #pragma unroll
      for (int c = 0; c < 8; ++c)
        f[c] = tl[idx[c]];
    }

    float o0 = 0.0f, o1 = 0.0f;
    #pragma unroll
    for (int c = 0; c < 8; ++c) {
      o0 = fmaf(w[c], f[c].x, o0);
      o1 = fmaf(w[c], f[c].y, o1);
    }
    acc[2 * l + 0] = o0;
    acc[2 * l + 1] = o1;
  }

  // 128 B contiguous per point; NT stores keep the streaming output from
  // evicting the L2-resident hash tables
  v4f* op = (v4f*)(out + (size_t)n * 32u);
  #pragma unroll
  for (int q = 0; q < 8; ++q) {
    v4f v = { acc[4 * q + 0], acc[4 * q + 1], acc[4 * q + 2], acc[4 * q + 3] };
    __builtin_nontemporal_store(v, op + q);
  }
}

extern "C" void kernel_launch(void* const* d_in, const int* in_sizes, int n_in,
                              void* d_out, int out_size, void* d_ws, size_t ws_size,
                              hipStream_t stream) {
  const float* x     = (const float*)d_in[0];
  const float* table = (const float*)d_in[1];
  float* out         = (float*)d_out;
  int npts = in_sizes[0] / 3;

  const int block = 256;
  const int grid  = (npts + block - 1) / block;
  hashgrid_fwd<<<grid, block, 0, stream>>>(x, table, out, npts);
}